// ProxyDeltaLinear_78245714199343
// MI455X (gfx1250) — compile-verified
//
#include <hip/hip_runtime.h>
#include <cstdint>
#include <cstddef>

// ---------------------------------------------------------------------------
// Forward of the reference reduces exactly to:
//   out[n,m] = (0.5 - mean_d |tanh(x[n,d]) - tanh(w[m,d])|) * gain[m]
// (the l2/GEMM branch is gradient-only: stop_gradient(l1-l2)+l2 == l1 in fwd)
// VALU-bound problem (~2.1e9 f32 ops vs ~1.7us memory floor on MI455X), so:
//  - tanh precomputed once into workspace (pass 1)
//  - tiled L1-distance kernel, 128x64 block tile, 8x4 register tile,
//    K staged in 32-deep chunks, double-buffered with CDNA5 async-to-LDS
//    (global_load_async_to_lds_b128 + s_wait_asynccnt) to overlap copy/compute
// ---------------------------------------------------------------------------

namespace {
constexpr int kDin    = 128;
constexpr int kDout   = 512;
constexpr int kRows   = 16384;            // B*T
constexpr int kBN     = 128;              // rows per block
constexpr int kBM     = 64;               // cols per block
constexpr int kCD     = 32;               // K-chunk depth
constexpr int kNChunk = kDin / kCD;       // 4
constexpr int kTPB    = 256;              // 8 wave32
constexpr int kXStr   = kCD;              // x rows broadcast-read: no pad needed
constexpr int kWStr   = kCD + 4;          // 36 dwords -> conflict-free strided b128
constexpr int kXElems = kBN * kXStr;      // 4096 floats (16 KB)
constexpr int kWElems = kBM * kWStr;      // 2304 floats (9 KB)
} // namespace

// Low 32 bits of a generic pointer to LDS are the DS-relative byte offset
// (AS3 offset) on AMDGPU -- exactly what the async-to-LDS VDST expects.
__device__ __forceinline__ uint32_t lds_byte_off(const void* p) {
  return (uint32_t)(uintptr_t)p;
}

// CDNA5 async copy: 16 bytes per lane, global -> LDS, tracked by ASYNCcnt.
__device__ __forceinline__ void async_cp_b128(const float* gsrc, const float* ldst) {
  uint32_t loff  = lds_byte_off(ldst);
  uint64_t gaddr = (uint64_t)(uintptr_t)gsrc;
  asm volatile("global_load_async_to_lds_b128 %0, %1, off"
               :: "v"(loff), "v"(gaddr)
               : "memory");
}

__device__ __forceinline__ void wait_asynccnt6() {
#if __has_builtin(__builtin_amdgcn_s_wait_asynccnt)
  __builtin_amdgcn_s_wait_asynccnt(6);
#else
  asm volatile("s_wait_asynccnt 0x6" ::: "memory");
#endif
}
__device__ __forceinline__ void wait_asynccnt0() {
#if __has_builtin(__builtin_amdgcn_s_wait_asynccnt)
  __builtin_amdgcn_s_wait_asynccnt(0);
#else
  asm volatile("s_wait_asynccnt 0x0" ::: "memory");
#endif
}

// ---------------- pass 1: elementwise tanh into workspace -------------------
__global__ __launch_bounds__(256) void act_kernel(const float* __restrict__ x,
                                                  const float* __restrict__ w,
                                                  float* __restrict__ xa,
                                                  float* __restrict__ wa) {
  constexpr int kXN4 = (kRows * kDin) / 4;   // 524288 float4
  constexpr int kWN4 = (kDout * kDin) / 4;   // 16384 float4
  int idx    = (int)(blockIdx.x * blockDim.x + threadIdx.x);
  int stride = (int)(gridDim.x * blockDim.x);
  for (int i = idx; i < kXN4 + kWN4; i += stride) {
    const float4* src;
    float4*       dst;
    if (i < kXN4) { src = (const float4*)x + i;         dst = (float4*)xa + i; }
    else          { src = (const float4*)w + (i - kXN4); dst = (float4*)wa + (i - kXN4); }
    float4 v = *src;
    v.x = tanhf(v.x); v.y = tanhf(v.y); v.z = tanhf(v.z); v.w = tanhf(v.w);
    *dst = v;
  }
}

// ---------------- tile loaders ---------------------------------------------
// Per chunk per thread: 4 x-float4 (128x32 tile) + 2 w-float4 (64x32 tile).
__device__ __forceinline__ void load_chunk_async(const float* __restrict__ xg,
                                                 const float* __restrict__ wg,
                                                 float* sx, float* sw,
                                                 int n0, int m0, int d0, int tid) {
#pragma unroll
  for (int u = 0; u < 4; ++u) {
    int idx = tid + u * kTPB;           // 0..1023
    int row = idx >> 3;                 // 0..127
    int dc  = (idx & 7) << 2;           // 0,4,...,28
    async_cp_b128(xg + (size_t)(n0 + row) * kDin + d0 + dc, sx + row * kXStr + dc);
  }
#pragma unroll
  for (int u = 0; u < 2; ++u) {
    int idx = tid + u * kTPB;           // 0..511
    int row = idx >> 3;                 // 0..63
    int dc  = (idx & 7) << 2;
    async_cp_b128(wg + (size_t)(m0 + row) * kDin + d0 + dc, sw + row * kWStr + dc);
  }
}

template <bool ACT>
__device__ __forceinline__ void load_chunk_sync(const float* __restrict__ xg,
                                                const float* __restrict__ wg,
                                                float* sx, float* sw,
                                                int n0, int m0, int d0, int tid) {
#pragma unroll
  for (int u = 0; u < 4; ++u) {
    int idx = tid + u * kTPB;
    int row = idx >> 3;
    int dc  = (idx & 7) << 2;
    float4 v = *(const float4*)(xg + (size_t)(n0 + row) * kDin + d0 + dc);
    if (ACT) { v.x = tanhf(v.x); v.y = tanhf(v.y); v.z = tanhf(v.z); v.w = tanhf(v.w); }
    *(float4*)(sx + row * kXStr + dc) = v;
  }
#pragma unroll
  for (int u = 0; u < 2; ++u) {
    int idx = tid + u * kTPB;
    int row = idx >> 3;
    int dc  = (idx & 7) << 2;
    float4 v = *(const float4*)(wg + (size_t)(m0 + row) * kDin + d0 + dc);
    if (ACT) { v.x = tanhf(v.x); v.y = tanhf(v.y); v.z = tanhf(v.z); v.w = tanhf(v.w); }
    *(float4*)(sw + row * kWStr + dc) = v;
  }
}

// ---------------- pass 2: L1-cdist + epilogue ------------------------------
// PRE = true : xg/wg are pre-activated (workspace); uses async-to-LDS pipeline
// PRE = false: xg/wg are raw inputs; tanh applied inline while staging
template <bool PRE>
__global__ __launch_bounds__(kTPB) void l1_kernel(const float* __restrict__ xg,
                                                  const float* __restrict__ wg,
                                                  const float* __restrict__ gain,
                                                  float* __restrict__ out) {
  __shared__ __align__(16) float sx[2][kXElems];
  __shared__ __align__(16) float sw[2][kWElems];

  const int tid = (int)threadIdx.x;
  const int tx  = tid & 15;             // column group
  const int ty  = tid >> 4;             // row group
  const int m0  = (int)blockIdx.x * kBM;
  const int n0  = (int)blockIdx.y * kBN;

  float acc[8][4];
#pragma unroll
  for (int i = 0; i < 8; ++i)
#pragma unroll
    for (int j = 0; j < 4; ++j) acc[i][j] = 0.0f;

  if constexpr (PRE) load_chunk_async(xg, wg, sx[0], sw[0], n0, m0, 0, tid);
  else               load_chunk_sync<true>(xg, wg, sx[0], sw[0], n0, m0, 0, tid);

#pragma unroll
  for (int k = 0; k < kNChunk; ++k) {
    const int cur = k & 1;
    const int nxt = cur ^ 1;
    if (k + 1 < kNChunk) {
      // buf[nxt] was last read in iteration k-1; trailing barrier guarantees safety
      if constexpr (PRE) load_chunk_async(xg, wg, sx[nxt], sw[nxt], n0, m0, (k + 1) * kCD, tid);
      else               load_chunk_sync<true>(xg, wg, sx[nxt], sw[nxt], n0, m0, (k + 1) * kCD, tid);
    }
    if constexpr (PRE) {
      // async loads complete in order: <=6 outstanding means chunk k has landed
      if (k + 1 < kNChunk) wait_asynccnt6();
      else                 wait_asynccnt0();
    }
    __syncthreads();

    const float* xb = sx[cur];
    const float* wb = sw[cur];
#pragma unroll
    for (int d = 0; d < kCD; d += 4) {
      float4 xv[8];
      float4 wv[4];
#pragma unroll
      for (int i = 0; i < 8; ++i)      // wave-broadcast reads, conflict-free
        xv[i] = *(const float4*)(xb + (ty + i * 16) * kXStr + d);
#pragma unroll
      for (int j = 0; j < 4; ++j)      // stride-36 rows -> all 64 banks covered
        wv[j] = *(const float4*)(wb + (tx + j * 16) * kWStr + d);
#pragma unroll
      for (int i = 0; i < 8; ++i)
#pragma unroll
        for (int j = 0; j < 4; ++j) {
          float s0 = __builtin_fabsf(xv[i].x - wv[j].x) + __builtin_fabsf(xv[i].y - wv[j].y);
          float s1 = __builtin_fabsf(xv[i].z - wv[j].z) + __builtin_fabsf(xv[i].w - wv[j].w);
          acc[i][j] += s0 + s1;
        }
    }
    __syncthreads();
  }

  // epilogue: out = (0.5 - l1_mean) * gain
  const float inv = 1.0f / (float)kDin;
  float g[4];
#pragma unroll
  for (int j = 0; j < 4; ++j) g[j] = gain[m0 + tx + j * 16];
#pragma unroll
  for (int i = 0; i < 8; ++i) {
    const size_t n = (size_t)(n0 + ty + i * 16);
#pragma unroll
    for (int j = 0; j < 4; ++j) {
      const int m = m0 + tx + j * 16;
      out[n * kDout + m] = (0.5f - acc[i][j] * inv) * g[j];
    }
  }
}

// ---------------------------------------------------------------------------
extern "C" void kernel_launch(void* const* d_in, const int* in_sizes, int n_in,
                              void* d_out, int out_size, void* d_ws, size_t ws_size,
                              hipStream_t stream) {
  (void)in_sizes; (void)n_in; (void)out_size;
  const float* x    = (const float*)d_in[0];   // [4,4096,128] f32
  const float* w    = (const float*)d_in[1];   // [512,128]    f32
  const float* gain = (const float*)d_in[2];   // [1,512,1]    f32
  float*       out  = (float*)d_out;           // [4,4096,512] f32

  const size_t need = (size_t)(kRows * kDin + kDout * kDin) * sizeof(float);
  dim3 grid(kDout / kBM, kRows / kBN);         // (8, 128)

  if (d_ws != nullptr && ws_size >= need) {
    float* xa = (float*)d_ws;
    float* wa = xa + (size_t)kRows * kDin;
    const int totalV4 = (kRows * kDin + kDout * kDin) / 4;   // 540672
    act_kernel<<<(totalV4 + 255) / 256, 256, 0, stream>>>(x, w, xa, wa);
    l1_kernel<true><<<grid, kTPB, 0, stream>>>(xa, wa, gain, out);
  } else {
    l1_kernel<false><<<grid, kTPB, 0, stream>>>(x, w, gain, out);
  }
}